// TRELLIS_68573447848293
// MI455X (gfx1250) — compile-verified
//
#include <hip/hip_runtime.h>
#include <hip/hip_bf16.h>

// MI455X / gfx1250, wave32. FP32 WMMA 16x16x4 path.
//
// Problem: N=4096, D=128 telescoping-density-ratio loss -> 9 scalars.
//   logits = -log1p(||y_i - x_j||^2)  (Cauchy kernel, scale=1)
//   pass 1: GEMM + one-pass sums (exp(logit)=1/(1+sq), logits<=0 so shift=0 LSE is exact)
//   pass 2: GEMM recompute + sigmoid statistics that depend on global baselines
//   finalize: per-row logmeanexp for classifier TP, assemble 9 outputs.

#define TRELLIS_N 4096
#define TRELLIS_D 128

typedef float v2f __attribute__((ext_vector_type(2)));
typedef float v8f __attribute__((ext_vector_type(8)));

__device__ __forceinline__ float trellis_sigmoid(float x) {
    return 1.0f / (1.0f + __expf(-x));
}

// ---- GEMM core: one wave computes a 16x64 strip (4 x 16x16 WMMA tiles) of y @ x^T ----
// A = z_y rows (M dim), B = z_x rows as columns (N dim). K = 128 in steps of 4.
__device__ __forceinline__ void trellis_gemm_strip(const float* __restrict__ zy,
                                                   const float* __restrict__ zx,
                                                   int ti, int tjb, int lane,
                                                   v8f c[4]) {
    const int frow = lane & 15;          // A row / B column within tile
    const int koff = (lane >> 4) << 1;   // k sub-offset {0,2} per lane half
    const float* arow = zy + (ti * 16 + frow) * TRELLIS_D;
    const float* brow0 = zx + ((tjb + 0) * 16 + frow) * TRELLIS_D;
    const float* brow1 = zx + ((tjb + 1) * 16 + frow) * TRELLIS_D;
    const float* brow2 = zx + ((tjb + 2) * 16 + frow) * TRELLIS_D;
    const float* brow3 = zx + ((tjb + 3) * 16 + frow) * TRELLIS_D;
#pragma unroll 8
    for (int k = 0; k < TRELLIS_D; k += 4) {
        v2f a  = *(const v2f*)(arow  + k + koff);
        v2f b0 = *(const v2f*)(brow0 + k + koff);
        v2f b1 = *(const v2f*)(brow1 + k + koff);
        v2f b2 = *(const v2f*)(brow2 + k + koff);
        v2f b3 = *(const v2f*)(brow3 + k + koff);
        c[0] = __builtin_amdgcn_wmma_f32_16x16x4_f32(false, a, false, b0, (short)0, c[0], false, false);
        c[1] = __builtin_amdgcn_wmma_f32_16x16x4_f32(false, a, false, b1, (short)0, c[1], false, false);
        c[2] = __builtin_amdgcn_wmma_f32_16x16x4_f32(false, a, false, b2, (short)0, c[2], false, false);
        c[3] = __builtin_amdgcn_wmma_f32_16x16x4_f32(false, a, false, b3, (short)0, c[3], false, false);
    }
}

// ---- row squared norms: one wave per row, 8 rows per block ----
__global__ __launch_bounds__(256) void trellis_norms(const float* __restrict__ src,
                                                     float* __restrict__ dst) {
    const int lane = threadIdx.x & 31;
    const int row = blockIdx.x * 8 + (threadIdx.x >> 5);
    const float* p = src + row * TRELLIS_D;
    float s = 0.0f;
#pragma unroll
    for (int k = 0; k < TRELLIS_D; k += 32) {
        float v = p[k + lane];
        s += v * v;
    }
#pragma unroll
    for (int m = 16; m > 0; m >>= 1) s += __shfl_xor(s, m, 32);
    if (lane == 0) dst[row] = s;
}

__global__ __launch_bounds__(256) void trellis_zero(float* __restrict__ p, int n) {
    int i = blockIdx.x * 256 + threadIdx.x;
    if (i < n) p[i] = 0.0f;
}

// ---- pass 1: GEMM -> logits; accumulate S_negexp, S_neg, S_posexp, S_pos,
//      per-row sum(exp(neg)), and store pos_logits[i]. ----
__global__ __launch_bounds__(32) void trellis_pass1(const float* __restrict__ zx,
                                                    const float* __restrict__ zy,
                                                    const float* __restrict__ nx,
                                                    const float* __restrict__ ny,
                                                    float* __restrict__ acc,
                                                    float* __restrict__ rowsumexp,
                                                    float* __restrict__ pos) {
    const int lane = threadIdx.x;
    const int ti = blockIdx.x;
    const int tjb = blockIdx.y * 4;
    v8f c[4] = {};
    trellis_gemm_strip(zy, zx, ti, tjb, lane, c);

    const int half = lane >> 4;
    const int col = lane & 15;
    float nyv[8];
#pragma unroll
    for (int r = 0; r < 8; ++r) nyv[r] = ny[ti * 16 + r + half * 8];

    float negexp_l = 0.0f, neg_l = 0.0f;
    float rowacc[8] = {};
#pragma unroll
    for (int t = 0; t < 4; ++t) {
        const int gj = (tjb + t) * 16 + col;
        const float nxv = nx[gj];
#pragma unroll
        for (int r = 0; r < 8; ++r) {
            const int gi = ti * 16 + r + half * 8;
            float sq = fmaxf(nyv[r] + nxv - 2.0f * c[t][r], 0.0f);
            float lg = -log1pf(sq);
            float e = 1.0f / (1.0f + sq);   // == exp(lg), exact
            if (gi == gj) {                  // diagonal: positive logit
                pos[gi] = lg;
                atomicAdd(&acc[2], e);
                atomicAdd(&acc[3], lg);
            } else {
                negexp_l += e;
                neg_l += lg;
                rowacc[r] += e;
            }
        }
    }
#pragma unroll
    for (int m = 16; m > 0; m >>= 1) {
        negexp_l += __shfl_xor(negexp_l, m, 32);
        neg_l    += __shfl_xor(neg_l, m, 32);
    }
    if (lane == 0) {
        atomicAdd(&acc[0], negexp_l);
        atomicAdd(&acc[1], neg_l);
    }
#pragma unroll
    for (int r = 0; r < 8; ++r) {
        float v = rowacc[r];
#pragma unroll
        for (int m = 8; m > 0; m >>= 1) v += __shfl_xor(v, m, 32);  // reduce 16 cols per half
        if (col == 0) atomicAdd(&rowsumexp[ti * 16 + r + half * 8], v);
    }
}

// ---- pass 2: GEMM recompute -> sum(way), sum(sigmoid(gpos)), sum(sigmoid(gneg)) ----
__global__ __launch_bounds__(32) void trellis_pass2(const float* __restrict__ zx,
                                                    const float* __restrict__ zy,
                                                    const float* __restrict__ nx,
                                                    const float* __restrict__ ny,
                                                    const float* __restrict__ pos,
                                                    float* __restrict__ acc) {
    const int lane = threadIdx.x;
    const int ti = blockIdx.x;
    const int tjb = blockIdx.y * 4;
    v8f c[4] = {};
    trellis_gemm_strip(zy, zx, ti, tjb, lane, c);

    const float Mf = (float)TRELLIS_N * (float)(TRELLIS_N - 1);
    const float S_negexp = acc[0];
    const float S_posexp = acc[2];
    const float lb1 = __logf(S_negexp) - __logf(Mf);
    const float lb0 = __logf(0.5f * ((float)(TRELLIS_N - 1) * S_posexp + S_negexp)) - __logf(Mf);
    const float L = lb0 + lb1;

    const int half = lane >> 4;
    const int col = lane & 15;
    float nyv[8], posi[8], epos[8];
#pragma unroll
    for (int r = 0; r < 8; ++r) {
        const int gi = ti * 16 + r + half * 8;
        nyv[r] = ny[gi];
        posi[r] = pos[gi];
        epos[r] = __expf(posi[r]);
    }

    float way_l = 0.0f, sp_l = 0.0f, sn_l = 0.0f;
#pragma unroll
    for (int t = 0; t < 4; ++t) {
        const int gj = (tjb + t) * 16 + col;
        const float nxv = nx[gj];
#pragma unroll
        for (int r = 0; r < 8; ++r) {
            const int gi = ti * 16 + r + half * 8;
            if (gi == gj) continue;          // off-diagonal only
            float sq = fmaxf(nyv[r] + nxv - 2.0f * c[t][r], 0.0f);
            float lg = -log1pf(sq);
            float e = 1.0f / (1.0f + sq);    // exp(neg)
            float way = __logf(0.5f * (epos[r] + e));
            way_l += way;
            sp_l += trellis_sigmoid(posi[r] + way - L);  // sigmoid(gpos)
            sn_l += trellis_sigmoid(way + lg - L);       // sigmoid(gneg)
        }
    }
#pragma unroll
    for (int m = 16; m > 0; m >>= 1) {
        way_l += __shfl_xor(way_l, m, 32);
        sp_l  += __shfl_xor(sp_l, m, 32);
        sn_l  += __shfl_xor(sn_l, m, 32);
    }
    if (lane == 0) {
        atomicAdd(&acc[4], way_l);
        atomicAdd(&acc[5], sp_l);
        atomicAdd(&acc[6], sn_l);
    }
}

// ---- finalize: per-row logmeanexp for TP, assemble 9 outputs ----
__global__ __launch_bounds__(256) void trellis_finalize(const float* __restrict__ acc,
                                                        const float* __restrict__ pos,
                                                        const float* __restrict__ rowsumexp,
                                                        float* __restrict__ out) {
    __shared__ float sh[256];
    const int tid = threadIdx.x;
    const float Mf = (float)TRELLIS_N * (float)(TRELLIS_N - 1);
    const float Nm1 = (float)(TRELLIS_N - 1);
    const float S_negexp = acc[0];
    const float S_posexp = acc[2];
    const float lb1 = __logf(S_negexp) - __logf(Mf);
    const float lb0 = __logf(0.5f * (Nm1 * S_posexp + S_negexp)) - __logf(Mf);
    const float L = lb0 + lb1;

    float tp = 0.0f;
    for (int i = tid; i < TRELLIS_N; i += 256) {
        float pi = pos[i];
        // logmeanexp_j(gpos[i,:]) = pos_i - L + log( sum_j exp(way_ij) / (N-1) )
        float mean_wayexp = 0.5f * (Nm1 * __expf(pi) + rowsumexp[i]) / Nm1;
        float pos_flat = pi - L + __logf(mean_wayexp);
        tp += trellis_sigmoid(pos_flat);
    }
    sh[tid] = tp;
    __syncthreads();
    for (int s = 128; s > 0; s >>= 1) {
        if (tid < s) sh[tid] += sh[tid + s];
        __syncthreads();
    }
    if (tid == 0) {
        const float TP = sh[0] / (float)TRELLIS_N;
        const float way_mean = acc[4] / Mf;
        const float neg_mean = acc[1] / Mf;
        const float pos_mean = acc[3] / (float)TRELLIS_N;
        const float gpos_mean = pos_mean + way_mean - L;
        const float gneg_mean = way_mean + neg_mean - L;
        const float FP = acc[6] / Mf;
        out[0] = gpos_mean;
        out[1] = gneg_mean;
        out[2] = acc[5] / Mf;                 // sigmoid(gpos).mean()
        out[3] = FP;                          // sigmoid(gneg).mean()
        out[4] = (TP + (1.0f - FP)) * 0.5f;   // accuracy
        out[5] = TP;                          // recall
        out[6] = TP / (TP + FP);              // precision
        out[7] = L;                           // mean(gbase) = lb0 + lb1
        out[8] = 2.0f - gpos_mean;            // gatt.mean() + grep.mean() (grep.mean()==2 exactly)
    }
}

extern "C" void kernel_launch(void* const* d_in, const int* in_sizes, int n_in,
                              void* d_out, int out_size, void* d_ws, size_t ws_size,
                              hipStream_t stream) {
    (void)in_sizes; (void)n_in; (void)out_size; (void)ws_size;
    const float* zx = (const float*)d_in[0];   // z_x : [N, D]
    const float* zy = (const float*)d_in[1];   // z_y : [N, D]
    float* out = (float*)d_out;                // 9 floats

    // ws layout (floats): [acc 8][rowsumexp 4096][pos 4096][nx 4096][ny 4096]  (~64 KB)
    float* ws = (float*)d_ws;
    float* acc = ws;
    float* rowsumexp = ws + 8;
    float* pos = rowsumexp + TRELLIS_N;
    float* nx = pos + TRELLIS_N;
    float* ny = nx + TRELLIS_N;

    trellis_zero<<<(8 + TRELLIS_N + 255) / 256, 256, 0, stream>>>(ws, 8 + TRELLIS_N);
    trellis_norms<<<TRELLIS_N / 8, 256, 0, stream>>>(zx, nx);
    trellis_norms<<<TRELLIS_N / 8, 256, 0, stream>>>(zy, ny);

    dim3 grid(TRELLIS_N / 16, TRELLIS_N / 64);   // 256 x 64 waves, each owns a 16x64 strip
    trellis_pass1<<<grid, 32, 0, stream>>>(zx, zy, nx, ny, acc, rowsumexp, pos);
    trellis_pass2<<<grid, 32, 0, stream>>>(zx, zy, nx, ny, pos, acc);
    trellis_finalize<<<1, 256, 0, stream>>>(acc, pos, rowsumexp, out);
}